// LocalSelfAttentionBlock_25881472926456
// MI455X (gfx1250) — compile-verified
//
#include <hip/hip_runtime.h>

#define N_PTS 16384
#define DIM 256
#define HEADS 8
#define HD 32
#define KNN 20
#define MLP_HID 1024
#define LN_EPS 1e-5f
#define ATT_SCALE 0.17677669529663687f  // 1/sqrt(32)
#define KNN_CHUNK 4096
#define GEMM_KC 256                      // K-chunk staged in LDS (64 rows x 256 k = 64KB)

typedef __attribute__((ext_vector_type(2))) float v2f;
typedef __attribute__((ext_vector_type(8))) float v8f;

// ---------------------------------------------------------------------------
// kNN: positions scanned in 4096-point LDS chunks (64KB as padded float4 ->
// one ds_load_b128 broadcast per candidate). One thread per query; register
// top-20 insertion sort, stable (strict <) to match jax top_k lower-index
// tie break. Self excluded by j==q skip.
// ---------------------------------------------------------------------------
__global__ void knn_kernel(const float* __restrict__ pos, int* __restrict__ knn) {
  extern __shared__ float4 sp[];  // KNN_CHUNK float4 = 64KB
  const int tid = threadIdx.x;
  const int q = blockIdx.x * blockDim.x + tid;
  const float qx = pos[q * 3 + 0], qy = pos[q * 3 + 1], qz = pos[q * 3 + 2];
  float bd[KNN];
  int bi[KNN];
#pragma unroll
  for (int t = 0; t < KNN; ++t) { bd[t] = 3.4e38f; bi[t] = -1; }

  for (int base = 0; base < N_PTS; base += KNN_CHUNK) {
    __syncthreads();
    for (int i = tid; i < KNN_CHUNK; i += blockDim.x) {
      const int j = base + i;
      sp[i] = make_float4(pos[j * 3 + 0], pos[j * 3 + 1], pos[j * 3 + 2], 0.f);
    }
    __syncthreads();
#pragma unroll 2
    for (int i = 0; i < KNN_CHUNK; ++i) {
      const float4 p = sp[i];
      const float dx = p.x - qx, dy = p.y - qy, dz = p.z - qz;
      const float d = dx * dx + dy * dy + dz * dz;
      const int j = base + i;
      if (j == q) continue;
      if (d < bd[KNN - 1]) {
        bd[KNN - 1] = d;
        bi[KNN - 1] = j;
#pragma unroll
        for (int t = KNN - 1; t > 0; --t) {
          if (bd[t] < bd[t - 1]) {
            const float td = bd[t]; bd[t] = bd[t - 1]; bd[t - 1] = td;
            const int ti = bi[t];  bi[t] = bi[t - 1];  bi[t - 1] = ti;
          }
        }
      }
    }
  }
#pragma unroll
  for (int t = 0; t < KNN; ++t) knn[q * KNN + t] = bi[t];
}

// ---------------------------------------------------------------------------
// LayerNorm: one wave32 per row, 8 elements per lane, shfl_xor reduction.
// ---------------------------------------------------------------------------
__device__ inline float wave_sum(float v) {
#pragma unroll
  for (int m = 16; m > 0; m >>= 1) v += __shfl_xor(v, m, 32);
  return v;
}

__global__ void ln_kernel(const float* __restrict__ x, const float* __restrict__ w,
                          const float* __restrict__ b, float* __restrict__ h) {
  const int wave = threadIdx.x >> 5;
  const int lane = threadIdx.x & 31;
  const int row = blockIdx.x * (blockDim.x >> 5) + wave;
  const float* xr = x + (size_t)row * DIM;
  float vals[8];
  float s1 = 0.f, s2 = 0.f;
#pragma unroll
  for (int i = 0; i < 8; ++i) {
    const float v = xr[lane + 32 * i];
    vals[i] = v;
    s1 += v;
    s2 += v * v;
  }
  s1 = wave_sum(s1);
  s2 = wave_sum(s2);
  const float m = s1 * (1.0f / DIM);
  const float var = s2 * (1.0f / DIM) - m * m;
  const float rs = rsqrtf(var + LN_EPS);
  float* hr = h + (size_t)row * DIM;
#pragma unroll
  for (int i = 0; i < 8; ++i) {
    const int c = lane + 32 * i;
    hr[c] = (vals[i] - m) * rs * w[c] + b[c];
  }
}

// ---------------------------------------------------------------------------
// Weight transpose: Wt[k*Nc + c] = W[c*Kd + k]  (W is [Nc, Kd] row-major).
// One-time; makes GEMM B-fragment loads contiguous.
// ---------------------------------------------------------------------------
__global__ void transpose_kernel(const float* __restrict__ W, float* __restrict__ Wt,
                                 int Nc, int Kd) {
  const int i = blockIdx.x * blockDim.x + threadIdx.x;
  const int k = i / Nc;
  const int c = i - k * Nc;
  Wt[i] = W[(size_t)c * Kd + k];
}

// ---------------------------------------------------------------------------
// GEMM via full-precision CDNA5 WMMA: C[M,Nc] = A[M,Kd] @ W^T + bias (W given
// pre-transposed as Wt[Kd,Nc]), optional exact GELU, optional residual add.
// Block = 4 waves covering a 64-row x 128-col output tile. The A stripe is
// staged in LDS in 64x256 k-chunks (64KB). Each wave owns a 4x2 grid of
// 16x16 tiles (64 accumulator VGPRs -- fits the 256-VGPR window, no spills):
// per K-step, 4 ds_load_b64 (A) + 4 global_load_b32 (B) feed EIGHT
// independent v_wmma_f32_16x16x4_f32 accumulate chains.
// Fragment layouts per ISA 7.12.2:
//   A (16x4 f32): lane r=lane&15, VGPRs hold K = k+2*(lane>>4)+{0,1}
//   B (4x16 f32): col c=lane&15,  VGPRs hold K = k+2*(lane>>4)+{0,1}
//   C/D        : row = i + 8*(lane>>4), col = lane&15
// ---------------------------------------------------------------------------
__global__ void gemm_wmma_kernel(const float* __restrict__ A, const float* __restrict__ Wt,
                                 const float* __restrict__ bias, const float* __restrict__ R,
                                 float* __restrict__ C, int Kd, int Nc, int gelu) {
  extern __shared__ float sa[];  // 64 * GEMM_KC floats = 64KB
  const int tid = threadIdx.x;
  const int row0 = blockIdx.y * 64;
  const int wave = tid >> 5;
  const int lane = tid & 31;
  const int r = lane & 15;
  const int hsel = lane >> 4;
  const int cbase = blockIdx.x * 128 + wave * 32 + r;  // col of tile (.,0)

  v8f acc[4][2];
#pragma unroll
  for (int rt = 0; rt < 4; ++rt)
#pragma unroll
    for (int t = 0; t < 2; ++t) acc[rt][t] = (v8f){};

  for (int kc = 0; kc < Kd; kc += GEMM_KC) {
    __syncthreads();
    // stage 64 x GEMM_KC A-chunk as float4 (both sides 16B aligned)
    const float* Ab = A + (size_t)row0 * Kd + kc;
    for (int i4 = tid; i4 < (64 * GEMM_KC) / 4; i4 += blockDim.x) {
      const int row = i4 >> 6;          // GEMM_KC/4 = 64 float4 per row
      const int c4 = i4 & 63;
      const float4 vv = *(const float4*)(Ab + (size_t)row * Kd + c4 * 4);
      *(float4*)(sa + row * GEMM_KC + c4 * 4) = vv;
    }
    __syncthreads();

    for (int k = 0; k < GEMM_KC; k += 4) {
      const int ko = k + 2 * hsel;
      v2f a[4];
#pragma unroll
      for (int rt = 0; rt < 4; ++rt) {
        a[rt].x = sa[(rt * 16 + r) * GEMM_KC + ko];
        a[rt].y = sa[(rt * 16 + r) * GEMM_KC + ko + 1];
      }
      const float* w0 = Wt + (size_t)(kc + ko) * Nc + cbase;
      const float* w1 = w0 + Nc;
      v2f b[2];
#pragma unroll
      for (int t = 0; t < 2; ++t) {
        b[t].x = w0[16 * t];
        b[t].y = w1[16 * t];
      }
#pragma unroll
      for (int rt = 0; rt < 4; ++rt)
#pragma unroll
        for (int t = 0; t < 2; ++t)
          acc[rt][t] = __builtin_amdgcn_wmma_f32_16x16x4_f32(
              false, a[rt], false, b[t], (short)0, acc[rt][t], false, false);
    }
  }

#pragma unroll
  for (int rt = 0; rt < 4; ++rt) {
#pragma unroll
    for (int t = 0; t < 2; ++t) {
      const int col = cbase + 16 * t;
      const float bc = bias[col];
#pragma unroll
      for (int i = 0; i < 8; ++i) {
        const int row = row0 + rt * 16 + i + 8 * hsel;
        float v = acc[rt][t][i] + bc;
        if (gelu) v = 0.5f * v * (1.0f + erff(v * 0.70710678118654752f));
        if (R) v += R[(size_t)row * Nc + col];
        C[(size_t)row * Nc + col] = v;
      }
    }
  }
}

// ---------------------------------------------------------------------------
// Local attention: one thread per (point, head). K=20 neighbors, hd=32.
// Fully unrolled so q/scores/acc stay in VGPRs.
// ---------------------------------------------------------------------------
__global__ void attn_kernel(const float* __restrict__ q, const float* __restrict__ k,
                            const float* __restrict__ v, const int* __restrict__ knn,
                            float* __restrict__ out) {
  const int gid = blockIdx.x * blockDim.x + threadIdx.x;
  const int n = gid >> 3;
  const int h = gid & 7;
  const float* qr = q + (size_t)n * DIM + h * HD;
  float qreg[HD];
#pragma unroll
  for (int d = 0; d < HD; ++d) qreg[d] = qr[d];
  const int* nb = knn + n * KNN;
  float sc[KNN];
#pragma unroll
  for (int j = 0; j < KNN; ++j) {
    const float* kr = k + (size_t)nb[j] * DIM + h * HD;
    float s = 0.f;
#pragma unroll
    for (int d = 0; d < HD; ++d) s += qreg[d] * kr[d];
    sc[j] = s * ATT_SCALE;
  }
  float mx = sc[0];
#pragma unroll
  for (int j = 1; j < KNN; ++j) mx = fmaxf(mx, sc[j]);
  float sum = 0.f;
#pragma unroll
  for (int j = 0; j < KNN; ++j) {
    sc[j] = expf(sc[j] - mx);
    sum += sc[j];
  }
  const float inv = 1.0f / sum;
  float acc[HD];
#pragma unroll
  for (int d = 0; d < HD; ++d) acc[d] = 0.f;
#pragma unroll
  for (int j = 0; j < KNN; ++j) {
    const float wgt = sc[j] * inv;
    const float* vr = v + (size_t)nb[j] * DIM + h * HD;
#pragma unroll
    for (int d = 0; d < HD; ++d) acc[d] += wgt * vr[d];
  }
  float* orow = out + (size_t)n * DIM + h * HD;
#pragma unroll
  for (int d = 0; d < HD; ++d) orow[d] = acc[d];
}

// ---------------------------------------------------------------------------
extern "C" void kernel_launch(void* const* d_in, const int* in_sizes, int n_in,
                              void* d_out, int out_size, void* d_ws, size_t ws_size,
                              hipStream_t stream) {
  (void)in_sizes; (void)n_in; (void)out_size; (void)ws_size;
  const float* x     = (const float*)d_in[0];
  const float* pos   = (const float*)d_in[1];
  const float* ln1_w = (const float*)d_in[2];
  const float* ln1_b = (const float*)d_in[3];
  const float* Wq    = (const float*)d_in[4];
  const float* bq    = (const float*)d_in[5];
  const float* Wk    = (const float*)d_in[6];
  const float* bk    = (const float*)d_in[7];
  const float* Wv    = (const float*)d_in[8];
  const float* bv    = (const float*)d_in[9];
  const float* Wo    = (const float*)d_in[10];
  const float* bo    = (const float*)d_in[11];
  const float* ln2_w = (const float*)d_in[12];
  const float* ln2_b = (const float*)d_in[13];
  const float* W1    = (const float*)d_in[14];
  const float* b1    = (const float*)d_in[15];
  const float* W2    = (const float*)d_in[16];
  const float* b2    = (const float*)d_in[17];
  float* out = (float*)d_out;

  // Workspace (~84 MB): knn | h | [q k v attn_out] (64MB, reused as MLP hidden)
  //                     | transposed weights (3MB)
  char* ws = (char*)d_ws;
  size_t off = 0;
  int* knn = (int*)(ws + off);
  off += (size_t)N_PTS * KNN * sizeof(int);
  off = (off + 255) & ~(size_t)255;
  float* hbuf = (float*)(ws + off); off += (size_t)N_PTS * DIM * sizeof(float);
  float* qb   = (float*)(ws + off); off += (size_t)N_PTS * DIM * sizeof(float);
  float* kb   = (float*)(ws + off); off += (size_t)N_PTS * DIM * sizeof(float);
  float* vb   = (float*)(ws + off); off += (size_t)N_PTS * DIM * sizeof(float);
  float* ab   = (float*)(ws + off); off += (size_t)N_PTS * DIM * sizeof(float);
  float* wtq  = (float*)(ws + off); off += (size_t)DIM * DIM * sizeof(float);
  float* wtk  = (float*)(ws + off); off += (size_t)DIM * DIM * sizeof(float);
  float* wtv  = (float*)(ws + off); off += (size_t)DIM * DIM * sizeof(float);
  float* wto  = (float*)(ws + off); off += (size_t)DIM * DIM * sizeof(float);
  float* wt1  = (float*)(ws + off); off += (size_t)DIM * MLP_HID * sizeof(float);
  float* wt2  = (float*)(ws + off); off += (size_t)MLP_HID * DIM * sizeof(float);
  float* tb   = qb;   // MLP hidden [N, 1024] reuses the contiguous q/k/v/attn 64MB
  float* x1   = out;  // residual-1 result lives in d_out (same-thread RMW later)

  const size_t lds_knn  = (size_t)KNN_CHUNK * sizeof(float4);   // 64KB
  const size_t lds_gemm = (size_t)64 * GEMM_KC * sizeof(float); // 64KB

  // 0) one-time weight transposes (coalesced writes)
  transpose_kernel<<<dim3((DIM * DIM) / 256), dim3(256), 0, stream>>>(Wq, wtq, DIM, DIM);
  transpose_kernel<<<dim3((DIM * DIM) / 256), dim3(256), 0, stream>>>(Wk, wtk, DIM, DIM);
  transpose_kernel<<<dim3((DIM * DIM) / 256), dim3(256), 0, stream>>>(Wv, wtv, DIM, DIM);
  transpose_kernel<<<dim3((DIM * DIM) / 256), dim3(256), 0, stream>>>(Wo, wto, DIM, DIM);
  transpose_kernel<<<dim3((DIM * MLP_HID) / 256), dim3(256), 0, stream>>>(W1, wt1, MLP_HID, DIM);
  transpose_kernel<<<dim3((MLP_HID * DIM) / 256), dim3(256), 0, stream>>>(W2, wt2, DIM, MLP_HID);
  // 1) kNN graph
  knn_kernel<<<dim3(N_PTS / 256), dim3(256), lds_knn, stream>>>(pos, knn);
  // 2) LN1
  ln_kernel<<<dim3(N_PTS / 8), dim3(256), 0, stream>>>(x, ln1_w, ln1_b, hbuf);
  // 3) Q, K, V projections (WMMA f32, 4x2 tiles/wave)
  gemm_wmma_kernel<<<dim3(DIM / 128, N_PTS / 64), dim3(128), lds_gemm, stream>>>(
      hbuf, wtq, bq, nullptr, qb, DIM, DIM, 0);
  gemm_wmma_kernel<<<dim3(DIM / 128, N_PTS / 64), dim3(128), lds_gemm, stream>>>(
      hbuf, wtk, bk, nullptr, kb, DIM, DIM, 0);
  gemm_wmma_kernel<<<dim3(DIM / 128, N_PTS / 64), dim3(128), lds_gemm, stream>>>(
      hbuf, wtv, bv, nullptr, vb, DIM, DIM, 0);
  // 4) local attention over kNN
  attn_kernel<<<dim3((N_PTS * HEADS) / 256), dim3(256), 0, stream>>>(qb, kb, vb, knn, ab);
  // 5) output projection + residual 1 -> x1 (in d_out)
  gemm_wmma_kernel<<<dim3(DIM / 128, N_PTS / 64), dim3(128), lds_gemm, stream>>>(
      ab, wto, bo, x, x1, DIM, DIM, 0);
  // 6) LN2
  ln_kernel<<<dim3(N_PTS / 8), dim3(256), 0, stream>>>(x1, ln2_w, ln2_b, hbuf);
  // 7) MLP up-projection + exact GELU
  gemm_wmma_kernel<<<dim3(MLP_HID / 128, N_PTS / 64), dim3(128), lds_gemm, stream>>>(
      hbuf, wt1, b1, nullptr, tb, DIM, MLP_HID, 1);
  // 8) MLP down-projection + residual 2 -> out
  gemm_wmma_kernel<<<dim3(DIM / 128, N_PTS / 64), dim3(128), lds_gemm, stream>>>(
      tb, wt2, b2, x1, out, MLP_HID, DIM, 0);
}